// DetectionTargetLayer_27487790694781
// MI455X (gfx1250) — compile-verified
//
#include <hip/hip_runtime.h>
#include <math.h>
#include <stdint.h>

// Problem constants (from reference setup_inputs)
#define NB   4
#define NP   2000
#define NG   20
#define IMH  800
#define IMW  800
#define MH   28
#define MW   28
#define IOU_THRESH 0.5f

// Flat float offsets into d_out (tuple concatenated in return order)
#define OUT_PROPS  0                       // (4,2000,4)  = 32000
#define OUT_LABELS (NB*NP*4)               // (4,2000)    =  8000
#define OUT_DELTAS (OUT_LABELS + NB*NP)    // (4,2000,4)  = 32000
#define OUT_MBOX   (OUT_DELTAS + NB*NP*4)  // (4,2000,4)  = 32000
#define OUT_MASKS  (OUT_MBOX   + NB*NP*4)  // (4,2000,28,28) = 6272000

// ---------------------------------------------------------------------------
// Tensor Data Mover staging (gfx1250): DMA a 1-D run of dwords global -> LDS.
// D# layout per CDNA5 ISA ch.8 (CDNA5 ISA).
// Toolchain probe: this lane's clang expects the 6-arg builtin
//   (u32x4 g0, i32x8 g1, i32x4 g2, i32x4 g3, i32x8 extra, i32 cpol)
// ---------------------------------------------------------------------------
#if defined(__has_builtin)
#if __has_builtin(__builtin_amdgcn_tensor_load_to_lds) && \
    __has_builtin(__builtin_amdgcn_s_wait_tensorcnt)
#define HAVE_TDM 1
#endif
#endif

#ifdef HAVE_TDM
typedef unsigned int u32x4 __attribute__((ext_vector_type(4)));
typedef int          i32x4 __attribute__((ext_vector_type(4)));
typedef int          i32x8 __attribute__((ext_vector_type(8)));

typedef __attribute__((address_space(3))) unsigned char lds_byte_t;

__device__ __forceinline__ uint32_t lds_offset_of(const void* p) {
    // ptrtoint of an LDS (addrspace 3) pointer == byte offset in the
    // workgroup's LDS allocation (same space DS / TDM lds_addr uses).
    return (uint32_t)(uintptr_t)(lds_byte_t*)p;
}

__device__ __forceinline__ void tdm_load_1d_dwords(uint32_t lds_off,
                                                   const void* gptr,
                                                   uint32_t nelem /* dwords */) {
    const uint64_t ga = (uint64_t)(uintptr_t)gptr;
    u32x4 g0;
    g0.x = 1u;                                     // count=1, user D#, no gather
    g0.y = lds_off;                                // lds_addr (bytes)
    g0.z = (uint32_t)ga;                           // global_addr[31:0]
    g0.w = ((uint32_t)(ga >> 32) & 0x01FFFFFFu)    // global_addr[56:32]
         | (2u << 30);                             // type = 2 ("image")
    i32x8 g1;
    g1[0] = (int)(2u << 16);                       // data_size=4B; mask/pad/iter=0
    g1[1] = (int)((nelem & 0xFFFFu) << 16);        // tensor_dim0[15:0]
    g1[2] = (int)(((nelem >> 16) & 0xFFFFu)        // tensor_dim0[31:16]
         | (1u << 16));                            // tensor_dim1 = 1
    g1[3] = (int)((nelem & 0xFFFFu) << 16);        // tile_dim0 = nelem (dim1 hi = 0)
    g1[4] = 0;                                     // tile_dim1/tile_dim2 unused
    g1[5] = (int)nelem;                            // tensor_dim0_stride[31:0]
    g1[6] = 0;                                     // stride0 hi / stride1 lo
    g1[7] = 0;                                     // stride1 hi
    const i32x4 z4 = {0, 0, 0, 0};                 // groups 2/3: <=2-D tensor
    const i32x8 z8 = {0, 0, 0, 0, 0, 0, 0, 0};
    __builtin_amdgcn_tensor_load_to_lds(g0, g1, z4, z4, z8, 0 /* cpol */);
}
#endif // HAVE_TDM

// ---------------------------------------------------------------------------
// Kernel 1: IoU match + labels/deltas/boxes. One thread per proposal.
// blockIdx.x = batch index so GT data stages into LDS once per block
// (via the Tensor Data Mover when available).
// ---------------------------------------------------------------------------
__global__ __launch_bounds__(256) void dtl_match_kernel(
    const float* __restrict__ proposals,   // (NB,NP,4)
    const float* __restrict__ gt_boxes,    // (NB,NG,4)
    const int*   __restrict__ gt_labels,   // (NB,NG)
    float*       __restrict__ out,         // full output buffer
    int*         __restrict__ match)       // (NB*NP) matched gt id, -1 if not pos
{
    __shared__ float sbox[NG * 4];
    __shared__ int   slab[NG];
    __shared__ float sarea[NG];

    const int b   = blockIdx.x;
    const int tid = threadIdx.x;

#ifdef HAVE_TDM
    if (tid < 32) {                          // wave 0 only: one DMA issue per block
        tdm_load_1d_dwords(lds_offset_of(sbox), gt_boxes + (size_t)b * NG * 4, NG * 4);
        tdm_load_1d_dwords(lds_offset_of(slab), gt_labels + (size_t)b * NG,    NG);
        __builtin_amdgcn_s_wait_tensorcnt(0);
    }
    __syncthreads();
#else
    if (tid < NG * 4) sbox[tid] = gt_boxes[b * NG * 4 + tid];
    if (tid < NG)     slab[tid] = gt_labels[b * NG + tid];
    __syncthreads();
#endif
    if (tid < NG) {
        const float gx1 = sbox[tid * 4 + 0], gy1 = sbox[tid * 4 + 1];
        const float gx2 = sbox[tid * 4 + 2], gy2 = sbox[tid * 4 + 3];
        sarea[tid] = (gx2 - gx1) * (gy2 - gy1);
    }
    __syncthreads();

    const int p = blockIdx.y * blockDim.x + tid;
    if (p >= NP) return;
    const int idx = b * NP + p;

    const float4 pb = ((const float4*)proposals)[idx];
    const float px1 = pb.x, py1 = pb.y, px2 = pb.z, py2 = pb.w;
    const float areaP = (px2 - px1) * (py2 - py1);

    float best  = -1.0f;
    int   bestg = 0;
#pragma unroll 5
    for (int g = 0; g < NG; ++g) {
        const float gx1 = sbox[4 * g + 0], gy1 = sbox[4 * g + 1];
        const float gx2 = sbox[4 * g + 2], gy2 = sbox[4 * g + 3];
        const float iw = fmaxf(fminf(px2, gx2) - fmaxf(px1, gx1), 0.0f);
        const float ih = fmaxf(fminf(py2, gy2) - fmaxf(py1, gy1), 0.0f);
        const float inter = iw * ih;
        const float iou = inter / (areaP + sarea[g] - inter);
        if (iou > best) { best = iou; bestg = g; }  // strict '>' keeps first max (jnp.argmax)
    }
    const bool pos = (best >= IOU_THRESH);

    // out0: proposals pass-through
    ((float4*)(out + OUT_PROPS))[idx] = pb;

    // out1: matched labels (int -> float in flat float output)
    out[OUT_LABELS + idx] = pos ? (float)slab[bestg] : 0.0f;

    // out2/out3: deltas + matched gt boxes
    const float gx1 = sbox[4 * bestg + 0], gy1 = sbox[4 * bestg + 1];
    const float gx2 = sbox[4 * bestg + 2], gy2 = sbox[4 * bestg + 3];
    const float pw = px2 - px1, ph = py2 - py1;
    const float pcx = px1 + 0.5f * pw, pcy = py1 + 0.5f * ph;
    const float gw = gx2 - gx1, gh = gy2 - gy1;
    const float gcx = gx1 + 0.5f * gw, gcy = gy1 + 0.5f * gh;
    float4 d, mb;
    if (pos) {
        d  = make_float4((gcx - pcx) / pw, (gcy - pcy) / ph, logf(gw / pw), logf(gh / ph));
        mb = make_float4(gx1, gy1, gx2, gy2);
    } else {
        d  = make_float4(0.f, 0.f, 0.f, 0.f);
        mb = d;
    }
    ((float4*)(out + OUT_DELTAS))[idx] = d;
    ((float4*)(out + OUT_MBOX))[idx]   = mb;

    match[idx] = pos ? bestg : -1;
}

// ---------------------------------------------------------------------------
// Kernel 2: crop + bilinear resize of matched GT mask -> 28x28.
// One 128-thread (4 wave32) block per proposal; axis coords staged in LDS;
// tap rows pre-touched via the gfx1250 prefetch path (global_prefetch_b8).
// ---------------------------------------------------------------------------
__global__ __launch_bounds__(128) void dtl_mask_kernel(
    const float* __restrict__ proposals,   // (NB,NP,4)
    const float* __restrict__ gt_masks,    // (NB,NG,800,800)
    const int*   __restrict__ match,       // (NB*NP)
    float*       __restrict__ outm)        // (NB,NP,28,28)
{
    const int idx = blockIdx.x;            // b*NP + p
    const int b   = idx / NP;
    const int tid = threadIdx.x;
    float* o = outm + (size_t)idx * (MH * MW);

    const int g = match[idx];

    // jnp.round == round-half-even == rintf; then clip to image
    const float4 pb = ((const float4*)proposals)[idx];
    const int x1 = min(max((int)rintf(pb.x), 0), IMW - 1);
    const int y1 = min(max((int)rintf(pb.y), 0), IMH - 1);
    const int x2 = min(max((int)rintf(pb.z), 0), IMW - 1);
    const int y2 = min(max((int)rintf(pb.w), 0), IMH - 1);
    const bool valid = (x2 > x1) && (y2 > y1);

    if (g < 0 || !valid) {
        // zero-fill (output buffer is poisoned; must write every element)
        float4* o4 = (float4*)o;
        const float4 z = make_float4(0.f, 0.f, 0.f, 0.f);
        for (int t = tid; t < (MH * MW) / 4; t += blockDim.x) o4[t] = z;
        return;
    }

    __shared__ int   sY0[MH], sY1[MH], sX0[MW], sX1[MW];
    __shared__ float sWY[MH], sWX[MW];

    const int in_h = max(y2 - y1 + 1, 1);
    const int in_w = max(x2 - x1 + 1, 1);

    if (tid < MH) {
        // t = (j+0.5)*in_h/28 - 0.5, clipped to [0, in_h-1]
        float t = ((float)tid + 0.5f) * (float)in_h / 28.0f - 0.5f;
        t = fminf(fmaxf(t, 0.0f), (float)(in_h - 1));
        const float f = floorf(t);
        const int i0 = (int)f;
        sWY[tid] = t - f;
        sY0[tid] = y1 + i0;
        sY1[tid] = y1 + min(i0 + 1, in_h - 1);
    } else if (tid >= 64 && tid < 64 + MW) {
        const int i = tid - 64;
        float t = ((float)i + 0.5f) * (float)in_w / 28.0f - 0.5f;
        t = fminf(fmaxf(t, 0.0f), (float)(in_w - 1));
        const float f = floorf(t);
        const int i0 = (int)f;
        sWX[i] = t - f;
        sX0[i] = x1 + i0;
        sX1[i] = x1 + min(i0 + 1, in_w - 1);
    }
    __syncthreads();

    const float* mbase = gt_masks + (size_t)(b * NG + g) * (IMH * IMW);

    // Pre-touch the 56 tap rows into the near cache (gfx1250 global_prefetch_b8).
    if (tid < MH) {
        const int xa = sX0[0];
        __builtin_prefetch(mbase + (size_t)sY0[tid] * IMW + xa, 0, 0);
        __builtin_prefetch(mbase + (size_t)sY1[tid] * IMW + xa, 0, 0);
    }

    for (int t = tid; t < MH * MW; t += blockDim.x) {
        const int j = t / MW;
        const int i = t - j * MW;
        const float* r0 = mbase + (size_t)sY0[j] * IMW;
        const float* r1 = mbase + (size_t)sY1[j] * IMW;
        const int   xa = sX0[i], xb = sX1[i];
        const float wx = sWX[i], wy = sWY[j];
        const float v00 = r0[xa], v01 = r0[xb];
        const float v10 = r1[xa], v11 = r1[xb];
        const float top = (1.0f - wx) * v00 + wx * v01;
        const float bot = (1.0f - wx) * v10 + wx * v11;
        o[t] = (1.0f - wy) * top + wy * bot;
    }
}

// ---------------------------------------------------------------------------
extern "C" void kernel_launch(void* const* d_in, const int* in_sizes, int n_in,
                              void* d_out, int out_size, void* d_ws, size_t ws_size,
                              hipStream_t stream) {
    const float* proposals = (const float*)d_in[0];
    const float* gt_boxes  = (const float*)d_in[1];
    const int*   gt_labels = (const int*)  d_in[2];
    const float* gt_masks  = (const float*)d_in[3];
    float* out = (float*)d_out;
    int*   match = (int*)d_ws;                 // NB*NP ints = 32 KB scratch

    dim3 g1(NB, (NP + 255) / 256);
    dtl_match_kernel<<<g1, 256, 0, stream>>>(proposals, gt_boxes, gt_labels, out, match);

    dtl_mask_kernel<<<dim3(NB * NP), 128, 0, stream>>>(proposals, gt_masks, match,
                                                       out + OUT_MASKS);
    (void)in_sizes; (void)n_in; (void)out_size; (void)ws_size;
}